// DistributedVolumeArgmin_28681791602996
// MI455X (gfx1250) — compile-verified
//
#include <hip/hip_runtime.h>
#include <hip/hip_bf16.h>

typedef __attribute__((ext_vector_type(2))) float v2f;
typedef __attribute__((ext_vector_type(8))) float v8f;

#define CC      128          // channels = K
#define HH      96
#define WW      128
#define HW      12288        // HH*WW = M = N
#define NTILE   32           // N columns per LDS tile (2 WMMA column-tiles)
#define NCHUNK  8            // split of the N sweep across workgroups
#define NCOLS   (HW / NCHUNK)        // 1536 columns per chunk
#define NTC     (NCOLS / NTILE)      // 48 tile iterations per chunk
#define MWG     128          // 8 waves * 16 rows
#define LDSF    ((CC / 2) * NTILE * 2)   // floats per LDS buffer (4096 = 16 KB)

typedef unsigned int       u32;
typedef unsigned long long u64;

// Order-preserving map float -> u32 (monotonic for all finite floats).
__device__ __forceinline__ u32 fkey(float f) {
    u32 b = __float_as_uint(f);
    return (b & 0x80000000u) ? ~b : (b | 0x80000000u);
}
__device__ __forceinline__ float fkey_inv(u32 k) {
    u32 b = (k & 0x80000000u) ? (k & 0x7FFFFFFFu) : ~k;
    return __uint_as_float(b);
}

// ---------------- init: zero the packed (key,idx) array in d_out ----------------
__global__ void init_packed(u64* __restrict__ pk) {
    const int m = blockIdx.x * blockDim.x + threadIdx.x;
    if (m < HW) pk[m] = 0ull;   // any real packed value is > 0
}

// ---------------- main: fused GEMM + per-chunk max/argmax + atomic combine ------
__global__ __launch_bounds__(256)
void corr_argmax_wmma_f32(const float* __restrict__ fl,
                          const float* __restrict__ fr,
                          u64* __restrict__ pk)
{
    __shared__ float lds[2][LDSF];   // double-buffered B tile, [k/2][n][2] pairs

    const int tid  = threadIdx.x;
    const int lane = tid & 31;
    const int wave = tid >> 5;
    const int half = lane >> 4;      // 0: lanes 0-15, 1: lanes 16-31
    const int nl   = lane & 15;
    const int mbase  = blockIdx.x * MWG + wave * 16;   // x-major: co-resident WGs
    const int nstart = blockIdx.y * NCOLS;             // share the same B chunk
    const int mlane  = mbase + nl;

    // ---- A panel: 16 rows x 128 K as 32 fragments (16x4 f32 layout) ----
    v2f afrag[CC / 4];
#pragma unroll
    for (int s = 0; s < CC / 4; ++s) {
        const int k = 4 * s + 2 * half;
        afrag[s].x = fl[(size_t)(k + 0) * HW + mlane];
        afrag[s].y = fl[(size_t)(k + 1) * HW + mlane];
    }

    float bestv[8];
    int   besti[8];
#pragma unroll
    for (int r = 0; r < 8; ++r) { bestv[r] = -3.402823466e38f; besti[r] = 0; }

    float4 pf[4];   // staging regs: 16 floats/thread = one 128x32 B tile per WG

    // Stage tile 0 of this chunk into buffer 0.
#pragma unroll
    for (int i = 0; i < 4; ++i) {
        const int j  = tid + 256 * i;
        const int k  = j >> 3;
        const int n0 = (j & 7) * 4;
        pf[i] = *(const float4*)(fr + (size_t)k * HW + nstart + n0);
    }
#pragma unroll
    for (int i = 0; i < 4; ++i) {
        const int j  = tid + 256 * i;
        const int k  = j >> 3;
        const int n0 = (j & 7) * 4;
        float* dst = &lds[0][(k >> 1) * (NTILE * 2) + (k & 1)];
        dst[(n0 + 0) * 2] = pf[i].x;
        dst[(n0 + 1) * 2] = pf[i].y;
        dst[(n0 + 2) * 2] = pf[i].z;
        dst[(n0 + 3) * 2] = pf[i].w;
    }

    for (int t = 0; t < NTC; ++t) {
        __syncthreads();   // buf[t&1] staged; buf[(t+1)&1] fully consumed

        // Prefetch tile t+1 (latency hidden by the 64 WMMAs below).
        const int nb_next = nstart + ((t + 1 < NTC) ? (t + 1) * NTILE : 0);
#pragma unroll
        for (int i = 0; i < 4; ++i) {
            const int j  = tid + 256 * i;
            const int k  = j >> 3;
            const int n0 = (j & 7) * 4;
            pf[i] = *(const float4*)(fr + (size_t)k * HW + nb_next + n0);
        }

        // ---- Two independent WMMA accumulator chains over K = 128 ----
        const float* buf = lds[t & 1];
        v8f c0 = {0.f, 0.f, 0.f, 0.f, 0.f, 0.f, 0.f, 0.f};
        v8f c1 = {0.f, 0.f, 0.f, 0.f, 0.f, 0.f, 0.f, 0.f};
#pragma unroll
        for (int s = 0; s < CC / 4; ++s) {
            const v2f b0 = *(const v2f*)&buf[((2 * s + half) * NTILE + nl) * 2];
            const v2f b1 = *(const v2f*)&buf[((2 * s + half) * NTILE + 16 + nl) * 2];
            c0 = __builtin_amdgcn_wmma_f32_16x16x4_f32(
                     false, afrag[s], false, b0, (short)0, c0, false, false);
            c1 = __builtin_amdgcn_wmma_f32_16x16x4_f32(
                     false, afrag[s], false, b1, (short)0, c1, false, false);
        }

        // Running max/argmax; c0 columns precede c1 (strict > keeps lowest idx).
        const int nbase = nstart + t * NTILE;
        const int n0g = nbase + nl;
        const int n1g = nbase + 16 + nl;
#pragma unroll
        for (int r = 0; r < 8; ++r) {
            const float v0 = c0[r];
            if (v0 > bestv[r]) { bestv[r] = v0; besti[r] = n0g; }
            const float v1 = c1[r];
            if (v1 > bestv[r]) { bestv[r] = v1; besti[r] = n1g; }
        }

        // Stage tile t+1 into the other buffer.
        float* obuf = lds[(t + 1) & 1];
#pragma unroll
        for (int i = 0; i < 4; ++i) {
            const int j  = tid + 256 * i;
            const int k  = j >> 3;
            const int n0 = (j & 7) * 4;
            float* dst = &obuf[(k >> 1) * (NTILE * 2) + (k & 1)];
            dst[(n0 + 0) * 2] = pf[i].x;
            dst[(n0 + 1) * 2] = pf[i].y;
            dst[(n0 + 2) * 2] = pf[i].z;
            dst[(n0 + 3) * 2] = pf[i].w;
        }
    }

    // ---- Cross-lane reduce within each 16-lane group (wave32 shuffles) ----
#pragma unroll
    for (int r = 0; r < 8; ++r) {
        float v = bestv[r];
        int   i = besti[r];
#pragma unroll
        for (int mask = 8; mask >= 1; mask >>= 1) {
            const float ov = __shfl_xor(v, mask, 32);
            const int   oi = __shfl_xor(i, mask, 32);
            if (ov > v || (ov == v && oi < i)) { v = ov; i = oi; }
        }
        bestv[r] = v;
        besti[r] = i;
    }

    // ---- Combine chunks: one u64 atomic max per row leader ----
    // packed = (key << 32) | (0xFFFFFFFF - idx): ties resolve to lowest idx.
    if (nl == 0) {
#pragma unroll
        for (int r = 0; r < 8; ++r) {
            const int m = mbase + r + 8 * half;
            const u64 packed = ((u64)fkey(bestv[r]) << 32)
                             | (u64)(0xFFFFFFFFu - (u32)besti[r]);
            atomicMax(&pk[m], packed);
        }
    }
}

// ---------------- decode: packed (key,idx) -> (flow u,v) + cost -----------------
__global__ void decode_flow(const int* __restrict__ psx,
                            const int* __restrict__ psy,
                            float* __restrict__ out)
{
    const int m = blockIdx.x * blockDim.x + threadIdx.x;
    if (m >= HW) return;
    u64* pk = (u64*)out;                 // packed[m] lives where flow[m] goes
    const u64 packed = pk[m];            // read before overwrite (same thread)
    const float cost = fkey_inv((u32)(packed >> 32));
    const int   idx  = (int)(0xFFFFFFFFu - (u32)(packed & 0xFFFFFFFFu));
    const int sx = psx[0], sy = psy[0];
    const int x = m & (WW - 1);
    const int y = m / WW;
    out[2 * m + 0]  = (float)(x - (idx % WW) * sx);   // u
    out[2 * m + 1]  = (float)(y - (idx / WW) * sy);   // v
    out[2 * HW + m] = cost;                           // flow_cost
}

extern "C" void kernel_launch(void* const* d_in, const int* in_sizes, int n_in,
                              void* d_out, int out_size, void* d_ws, size_t ws_size,
                              hipStream_t stream) {
    const float* feat_l = (const float*)d_in[0];
    const float* feat_r = (const float*)d_in[1];
    const int*   sx     = (const int*)d_in[2];
    const int*   sy     = (const int*)d_in[3];
    float*       out    = (float*)d_out;
    u64*         pk     = (u64*)d_out;   // flow region doubles as packed scratch

    init_packed<<<dim3(HW / 256), dim3(256), 0, stream>>>(pk);

    dim3 grid(HW / MWG, NCHUNK);   // 96 x 8 = 768 workgroups
    dim3 block(256);               // 8 wave32s
    corr_argmax_wmma_f32<<<grid, block, 0, stream>>>(feat_l, feat_r, pk);

    decode_flow<<<dim3(HW / 256), dim3(256), 0, stream>>>(sx, sy, out);
}